// DynamicGNN_38594576121977
// MI455X (gfx1250) — compile-verified
//
#include <hip/hip_runtime.h>

#define NN 100000   // nodes (16 * 6250)
#define NE 1600000  // edges
#define DD 64       // feature dim

typedef __attribute__((ext_vector_type(2))) float v2f;
typedef __attribute__((ext_vector_type(8))) float v8f;

// ---------- prologue: gcn_norm ----------
__global__ void k_deg_init(float* __restrict__ deg) {
    int i = blockIdx.x * blockDim.x + threadIdx.x;
    if (i < NN) deg[i] = 1.0f;  // self-loop weight
}

__global__ void k_deg_accum(const int* __restrict__ dst, const float* __restrict__ w,
                            float* __restrict__ deg) {
    int e = blockIdx.x * blockDim.x + threadIdx.x;
    if (e < NE) atomicAdd(&deg[dst[e]], w[e]);
}

__global__ void k_dinv(float* __restrict__ deg) {
    int i = blockIdx.x * blockDim.x + threadIdx.x;
    if (i < NN) {
        float d = deg[i];
        deg[i] = (d > 0.0f) ? rsqrtf(d) : 0.0f;
    }
}

__global__ void k_norm(const int* __restrict__ src, const int* __restrict__ dst,
                       const float* __restrict__ w, const float* __restrict__ dinv,
                       float* __restrict__ norm) {
    int e = blockIdx.x * blockDim.x + threadIdx.x;
    if (e < NE) norm[e] = dinv[src[e]] * w[e] * dinv[dst[e]];
}

// ---------- fused GEMM: H = act(Xin) @ W ; AGG = H * dinv^2 ----------
// act(x) = FUSE ? relu(x + b_prev) : x      (fuses previous layer's epilogue)
// One wave per 16x16 tile; 16x V_WMMA_F32_16X16X4_F32 over K=64.
// A layout (16x4 f32): lane holds M = lane&15, K = v + 2*(lane>>4)
// B layout (4x16 f32): lane holds N = lane&15, K = v + 2*(lane>>4)
// D layout (16x16 f32): vgpr v -> M = v + 8*(lane>>4), N = lane&15
template <int FUSE>
__global__ void __launch_bounds__(256) k_gemm(const float* __restrict__ X,
                                              const float* __restrict__ W,
                                              const float* __restrict__ bprev,
                                              const float* __restrict__ dinv,
                                              float* __restrict__ H,
                                              float* __restrict__ AGG) {
    __shared__ float Wl[DD * DD];  // 16 KB
    __shared__ float bl[DD];
    int t = threadIdx.x;
    {
        const float4* Wv = (const float4*)W;
        float4* Wlv = (float4*)Wl;
#pragma unroll
        for (int i = 0; i < 4; ++i) Wlv[t + i * 256] = Wv[t + i * 256];
    }
    if (FUSE) {
        if (t < DD) bl[t] = bprev[t];
    }
    __syncthreads();

    int wave = t >> 5, lane = t & 31;
    int tile = blockIdx.x * 8 + wave;   // 25000 tiles total
    int strip = tile >> 2;              // 16-row strip index
    int ct = tile & 3;                  // 16-col tile index
    if (strip >= NN / 16) return;       // wave-uniform (never taken with exact grid)

    int r = lane & 15;
    int h2 = (lane >> 4) << 1;          // 0 or 2
    const float* xrow = X + (size_t)(strip * 16 + r) * DD;
    const float* wcol = Wl + ct * 16 + r;

    v8f acc = {0.f, 0.f, 0.f, 0.f, 0.f, 0.f, 0.f, 0.f};
#pragma unroll
    for (int k0 = 0; k0 < DD; k0 += 4) {
        float ax = xrow[k0 + h2];
        float ay = xrow[k0 + h2 + 1];
        if (FUSE) {  // compile-time: apply previous layer's bias + relu on read
            ax = fmaxf(ax + bl[k0 + h2], 0.0f);
            ay = fmaxf(ay + bl[k0 + h2 + 1], 0.0f);
        }
        v2f a, b;
        a.x = ax; a.y = ay;
        b.x = wcol[(k0 + h2) * DD];
        b.y = wcol[(k0 + h2 + 1) * DD];
        acc = __builtin_amdgcn_wmma_f32_16x16x4_f32(false, a, false, b,
                                                    (short)0, acc, false, false);
    }

    // epilogue: store H and the self-loop-initialized aggregation buffer
    int mrow = strip * 16 + ((lane >> 4) << 3);
    int ccol = ct * 16 + r;
    float* hp = H + (size_t)mrow * DD + ccol;
    float* ap = AGG + (size_t)mrow * DD + ccol;
#pragma unroll
    for (int v = 0; v < 8; ++v) {
        float s = dinv[mrow + v];
        hp[v * DD] = acc[v];
        ap[v * DD] = acc[v] * (s * s);
    }
}

// ---------- AGG[dst] += H[src] * norm (16 threads/edge, float4 each) ----------
__global__ void k_scatter(const int* __restrict__ src, const int* __restrict__ dst,
                          const float* __restrict__ norm, const float* __restrict__ H,
                          float* __restrict__ agg) {
    int t = blockIdx.x * blockDim.x + threadIdx.x;  // NE*16 threads
    int e = t >> 4;
    if (e < NE) {
        int q = t & 15;
        float n = norm[e];
        int s = src[e];
        int d = dst[e];
        float4 h = ((const float4*)(H + (size_t)s * DD))[q];
        float* a = agg + (size_t)d * DD + q * 4;
        atomicAdd(a + 0, h.x * n);
        atomicAdd(a + 1, h.y * n);
        atomicAdd(a + 2, h.z * n);
        atomicAdd(a + 3, h.w * n);
    }
}

// ---------- final layer only: Y = AGG + b ----------
__global__ void k_finalize(const float* __restrict__ agg, const float* __restrict__ bias,
                           float* __restrict__ Y) {
    int t = blockIdx.x * blockDim.x + threadIdx.x;
    if (t < NN * DD) {
        Y[t] = agg[t] + bias[t & (DD - 1)];
    }
}

extern "C" void kernel_launch(void* const* d_in, const int* in_sizes, int n_in,
                              void* d_out, int out_size, void* d_ws, size_t ws_size,
                              hipStream_t stream) {
    const float* x  = (const float*)d_in[0];
    const int*   ei = (const int*)d_in[1];     // edge_index [2, NE]
    const float* ew = (const float*)d_in[2];
    const float* W1 = (const float*)d_in[3];
    const float* b1 = (const float*)d_in[4];
    const float* W2 = (const float*)d_in[5];
    const float* b2 = (const float*)d_in[6];
    const float* W3 = (const float*)d_in[7];
    const float* b3 = (const float*)d_in[8];
    const int* src = ei;
    const int* dst = ei + NE;
    float* out = (float*)d_out;

    float* ws   = (float*)d_ws;
    float* dinv = ws;                       // NN floats (deg -> rsqrt(deg))
    float* norm = ws + NN;                  // NE floats
    float* H    = ws + NN + NE;             // NN*DD floats
    float* AGG  = H + (size_t)NN * DD;      // NN*DD floats (d_out is the 2nd agg buffer)

    const int B = 256;
    const int GEMM_BLK = (NN / 16) * 4 / 8;  // 3125
    const int SCAT_BLK = NE * 16 / B;        // 100000

    k_deg_init <<<(NN + B - 1) / B, B, 0, stream>>>(dinv);
    k_deg_accum<<<(NE + B - 1) / B, B, 0, stream>>>(dst, ew, dinv);
    k_dinv     <<<(NN + B - 1) / B, B, 0, stream>>>(dinv);
    k_norm     <<<(NE + B - 1) / B, B, 0, stream>>>(src, dst, ew, dinv, norm);

    // Layer 1: in = x (no fused epilogue), agg -> AGG
    k_gemm<0><<<GEMM_BLK, B, 0, stream>>>(x, W1, nullptr, dinv, H, AGG);
    k_scatter<<<SCAT_BLK, B, 0, stream>>>(src, dst, norm, H, AGG);

    // Layer 2: in = relu(AGG + b1) fused, agg -> d_out
    k_gemm<1><<<GEMM_BLK, B, 0, stream>>>(AGG, W2, b1, dinv, H, out);
    k_scatter<<<SCAT_BLK, B, 0, stream>>>(src, dst, norm, H, out);

    // Layer 3: in = relu(d_out + b2) fused, agg -> AGG, then out = AGG + b3
    k_gemm<1><<<GEMM_BLK, B, 0, stream>>>(out, W3, b2, dinv, H, AGG);
    k_scatter<<<SCAT_BLK, B, 0, stream>>>(src, dst, norm, H, AGG);
    k_finalize<<<NN * DD / B, B, 0, stream>>>(AGG, b3, out);
}